// GCNLayer_68719476736450
// MI455X (gfx1250) — compile-verified
//
#include <hip/hip_runtime.h>
#include <math.h>

#define NNODES 100000
#define NEDGES 1000000
#define DIM    64
#define BN_EPS 1e-5f

typedef __attribute__((ext_vector_type(2))) float v2f;
typedef __attribute__((ext_vector_type(8))) float v8f;

// ---------------------------------------------------------------------------
// Init: h[i][d] = bias[d] (bias folded into the scatter target), deg = 0,
// stats = 0.
// ---------------------------------------------------------------------------
__global__ void k_init(float* __restrict__ h, float* __restrict__ deg,
                       float* __restrict__ stats, const float* __restrict__ bias) {
    long long idx    = (long long)blockIdx.x * blockDim.x + threadIdx.x;
    long long stride = (long long)gridDim.x * blockDim.x;
    const long long total = (long long)NNODES * DIM;
    for (long long i = idx; i < total; i += stride) h[i] = bias[i & 63];
    for (long long i = idx; i < NNODES; i += stride) deg[i] = 0.0f;
    if (idx < 128) stats[idx] = 0.0f;
}

// ---------------------------------------------------------------------------
// In-degree on target nodes (float counts -> exact for integer degrees).
// ---------------------------------------------------------------------------
__global__ void k_degree(const int* __restrict__ col, float* __restrict__ deg) {
    int e = blockIdx.x * blockDim.x + threadIdx.x;
    if (e < NEDGES) atomicAdd(&deg[col[e]], 1.0f);
}

__global__ void k_dinv(const float* __restrict__ deg, float* __restrict__ dinv) {
    int i = blockIdx.x * blockDim.x + threadIdx.x;
    if (i < NNODES) {
        float d = deg[i];
        dinv[i] = (d > 0.0f) ? rsqrtf(d) : 0.0f;
    }
}

// ---------------------------------------------------------------------------
// xw = x @ W via V_WMMA_F32_16X16X4_F32.
// One wave per 16x16 output tile; K=64 walked in 16 steps of 4.
// A (16x4 f32): lane L holds M=L&15, K = (L>>4)*2 + {0,1} in {a.x,a.y}.
// B (4x16 f32): lane L holds N=L&15, K = (L>>4)*2 + {0,1} in {b.x,b.y}.
// C/D (16x16 f32): VGPR r, lane L -> M = (L>>4)*8 + r, N = L&15.
// Jobs = (100000/16) M-tiles * 4 N-tiles = 25000 waves = 3125 blocks exactly.
// ---------------------------------------------------------------------------
__global__ void k_gemm(const float* __restrict__ x, const float* __restrict__ w,
                       float* __restrict__ xw) {
    int wave = (blockIdx.x * blockDim.x + threadIdx.x) >> 5;
    int lane = threadIdx.x & 31;
    int tileN = wave & 3;
    int tileM = wave >> 2;
    int m  = lane & 15;
    int kh = lane >> 4;              // 0 or 1 (upper half of wave)
    int arow = (tileM * 16 + m) * DIM;
    int ncol = tileN * 16 + m;       // lane's output column for B

    v8f c = {};
#pragma unroll
    for (int k0 = 0; k0 < DIM; k0 += 4) {
        int ka = k0 + kh * 2;
        v2f a, b;
        a.x = x[arow + ka];
        a.y = x[arow + ka + 1];
        b.x = w[ka * DIM + ncol];
        b.y = w[(ka + 1) * DIM + ncol];
        c = __builtin_amdgcn_wmma_f32_16x16x4_f32(
                /*neg_a=*/false, a, /*neg_b=*/false, b,
                /*c_mod=*/(short)0, c, /*reuse_a=*/false, /*reuse_b=*/false);
    }

    int outBase = (tileM * 16 + kh * 8) * DIM + tileN * 16 + (lane & 15);
#pragma unroll
    for (int r = 0; r < 8; ++r) xw[outBase + r * DIM] = c[r];
}

// ---------------------------------------------------------------------------
// Message passing: one wave per edge, each lane owns a float2 of the 64-wide
// feature row. Coalesced 256B gather of xw[row], scaled scatter-atomics into
// h[col]. Both xw and h are L2-resident (51 MB < 192 MB L2).
// ---------------------------------------------------------------------------
__global__ void k_edges(const int* __restrict__ row, const int* __restrict__ col,
                        const float* __restrict__ dinv, const float* __restrict__ xw,
                        float* __restrict__ h) {
    int wid  = (blockIdx.x * blockDim.x + threadIdx.x) >> 5;  // edge id
    int lane = threadIdx.x & 31;
    int r = row[wid];
    int c = col[wid];
    float nrm = dinv[r] * dinv[c];
    const float2 v = *(const float2*)(xw + (size_t)r * DIM + 2 * lane);
    float* dst = h + (size_t)c * DIM + 2 * lane;
    atomicAdd(dst,     v.x * nrm);
    atomicAdd(dst + 1, v.y * nrm);
}

// ---------------------------------------------------------------------------
// Per-channel sum / sumsq. Grid stride is a multiple of 64 so each thread's
// channel (i & 63) is fixed; LDS reduces 4 lanes per channel, then one pair
// of atomics per channel per block.
// ---------------------------------------------------------------------------
__global__ void k_stats(const float* __restrict__ h, float* __restrict__ stats) {
    const long long total  = (long long)NNODES * DIM;
    long long stride = (long long)gridDim.x * blockDim.x;   // multiple of 64
    long long idx    = (long long)blockIdx.x * blockDim.x + threadIdx.x;
    float s = 0.0f, s2 = 0.0f;
    for (long long i = idx; i < total; i += stride) {
        float v = h[i];
        s += v;
        s2 += v * v;
    }
    __shared__ float ls[256], ls2[256];
    ls[threadIdx.x]  = s;
    ls2[threadIdx.x] = s2;
    __syncthreads();
    if (threadIdx.x < 64) {
        int t = threadIdx.x;
        float a = ls[t]  + ls[t + 64]  + ls[t + 128]  + ls[t + 192];
        float b = ls2[t] + ls2[t + 64] + ls2[t + 128] + ls2[t + 192];
        atomicAdd(&stats[t],      a);
        atomicAdd(&stats[64 + t], b);
    }
}

// Fold mean/var + bn affine into per-channel scale/shift.
__global__ void k_bnprep(const float* __restrict__ stats,
                         const float* __restrict__ bnw, const float* __restrict__ bnb,
                         float* __restrict__ scale, float* __restrict__ shift) {
    int ch = threadIdx.x;
    if (ch < 64) {
        const float invN = 1.0f / (float)NNODES;
        float mu  = stats[ch] * invN;
        float var = stats[64 + ch] * invN - mu * mu;
        float s = rsqrtf(var + BN_EPS) * bnw[ch];
        scale[ch] = s;
        shift[ch] = bnb[ch] - mu * s;
    }
}

// out = x + relu(h*scale + shift)  (pure streaming pass)
__global__ void k_final(const float* __restrict__ x, const float* __restrict__ h,
                        const float* __restrict__ scale, const float* __restrict__ shift,
                        float* __restrict__ out) {
    const long long total  = (long long)NNODES * DIM;
    long long stride = (long long)gridDim.x * blockDim.x;
    for (long long i = (long long)blockIdx.x * blockDim.x + threadIdx.x;
         i < total; i += stride) {
        int ch = (int)(i & 63);
        float v = h[i] * scale[ch] + shift[ch];
        out[i] = x[i] + fmaxf(v, 0.0f);
    }
}

// ---------------------------------------------------------------------------
// Launch. Inputs: 0:x [N*64 f32], 1:weight [64*64 f32], 2:bias [64 f32],
// 3:bn_weight [64 f32], 4:bn_bias [64 f32], 5:edge_index [2*E int32].
// ---------------------------------------------------------------------------
extern "C" void kernel_launch(void* const* d_in, const int* in_sizes, int n_in,
                              void* d_out, int out_size, void* d_ws, size_t ws_size,
                              hipStream_t stream) {
    const float* x    = (const float*)d_in[0];
    const float* w    = (const float*)d_in[1];
    const float* bias = (const float*)d_in[2];
    const float* bnw  = (const float*)d_in[3];
    const float* bnb  = (const float*)d_in[4];
    const int*   ei   = (const int*)d_in[5];
    const int*   row  = ei;             // edge_index[0]
    const int*   col  = ei + NEDGES;    // edge_index[1]
    float* out = (float*)d_out;

    float* ws    = (float*)d_ws;
    float* xw    = ws;                                   // N*64
    float* h     = xw  + (size_t)NNODES * DIM;           // N*64
    float* deg   = h   + (size_t)NNODES * DIM;           // N
    float* dinv  = deg + NNODES;                         // N
    float* stats = dinv + NNODES;                        // 128
    float* scale = stats + 128;                          // 64
    float* shift = scale + 64;                           // 64

    k_init  <<<1024, 256, 0, stream>>>(h, deg, stats, bias);
    k_degree<<<(NEDGES + 255) / 256, 256, 0, stream>>>(col, deg);
    k_dinv  <<<(NNODES + 255) / 256, 256, 0, stream>>>(deg, dinv);
    k_gemm  <<<(NNODES / 16) * 4 / 8, 256, 0, stream>>>(x, w, xw);   // 3125 blocks
    k_edges <<<NEDGES / 8, 256, 0, stream>>>(row, col, dinv, xw, h); // 1 wave/edge
    k_stats <<<256, 256, 0, stream>>>(h, stats);
    k_bnprep<<<1, 64, 0, stream>>>(stats, bnw, bnb, scale, shift);
    k_final <<<4096, 256, 0, stream>>>(x, h, scale, shift, out);
}